// PostModel_68736656605809
// MI455X (gfx1250) — compile-verified
//
#include <hip/hip_runtime.h>
#include <math.h>

// ---------------- problem constants (match reference) ----------------
#define BATCH   16
#define HH      512
#define WW      512
#define CCH     3
#define RCH     7
#define ECH     8
#define NOBJ    15
#define KPAIR   3
#define KP_THR  0.5f
#define IOU_THR 0.5f
#define RESIZE  512.0f

constexpr int THR1   = 256;                 // threads per plane-scan block
constexpr int LISTN  = NOBJ;                // per-thread local top-15
constexpr int PLANES = BATCH * CCH;         // 48
constexpr int CPP    = THR1 * LISTN;        // 3840 candidates per plane

struct Cand { float s; unsigned p; };

// CDNA5 WMMA fragment types (wave32): 16x4 f32 A/B = 2 VGPRs, 16x16 f32 C/D = 8 VGPRs
typedef __attribute__((ext_vector_type(2))) float v2f;
typedef __attribute__((ext_vector_type(8))) float v8f;
// TDM descriptor SGPR groups
typedef unsigned int u32x4 __attribute__((ext_vector_type(4)));
typedef unsigned int u32x8 __attribute__((ext_vector_type(8)));

// =====================================================================
// Kernel 1: peak-keep + deterministic per-thread top-15 per (b,c) plane.
// Memory-bound: streams the 50MB heatmap once (≈2.2us at 23.3 TB/s).
// Uses global_prefetch_b8 (gfx1250) to keep the stream ahead of the ALU.
// =====================================================================
__global__ void k_peaks(const float* __restrict__ hm, Cand* __restrict__ cand) {
    const int plane = blockIdx.x;           // b*CCH + c
    const int b = plane / CCH, c = plane % CCH;
    const int t = threadIdx.x;

    float    ls[LISTN];
    unsigned lp[LISTN];
#pragma unroll
    for (int q = 0; q < LISTN; ++q) { ls[q] = 0.0f; lp[q] = 0u; }

    const size_t base = (size_t)b * HH * WW * CCH;
    const int NPIX = HH * WW;

    for (int p = t; p < NPIX; p += THR1) {
        const int y = p >> 9, x = p & (WW - 1);
        const float v = hm[base + (size_t)p * CCH + c];
        // gfx1250 prefetch (global_prefetch_b8) a few strides ahead
        if (p + THR1 * 8 < NPIX)
            __builtin_prefetch(hm + base + (size_t)(p + THR1 * 8) * CCH + c, 0, 1);

        // 3x3 SAME max-pool peak test: keep iff v >= every neighbor
        bool peak = true;
#pragma unroll
        for (int dy = -1; dy <= 1; ++dy) {
            const int yy = y + dy;
            if (yy < 0 || yy >= HH) continue;
#pragma unroll
            for (int dx = -1; dx <= 1; ++dx) {
                const int xx = x + dx;
                if (xx < 0 || xx >= WW) continue;
                peak = peak && (v >= hm[base + (size_t)(yy * WW + xx) * CCH + c]);
            }
        }
        if (!peak) continue;

        float s = v; unsigned pi = (unsigned)p;
        // gate: beats the current tail? (ties -> lower flat index, like jax top_k)
        if (s > ls[LISTN - 1] || (s == ls[LISTN - 1] && pi < lp[LISTN - 1])) {
#pragma unroll
            for (int q = 0; q < LISTN; ++q) {
                const bool bet = (s > ls[q]) || (s == ls[q] && pi < lp[q]);
                if (bet) { float ts = ls[q]; unsigned tp = lp[q];
                           ls[q] = s; lp[q] = pi; s = ts; pi = tp; }
            }
        }
    }

    Cand* out = cand + (size_t)plane * CPP + (size_t)t * LISTN;
#pragma unroll
    for (int q = 0; q < LISTN; ++q) { out[q].s = ls[q]; out[q].p = lp[q]; }
}

// =====================================================================
// Kernel 2: per plane — TDM DMA of the 30KB candidate block into LDS
// (tensor_load_to_lds + s_wait_tensorcnt), then global top-15, decode
// boxes, validity mask, NMS.
// =====================================================================
__device__ inline float iou_of(const float* a, const float* c) {
    const float areaA = (a[2] - a[0]) * (a[3] - a[1]);
    const float areaB = (c[2] - c[0]) * (c[3] - c[1]);
    const float ih = fmaxf(fminf(a[2], c[2]) - fmaxf(a[0], c[0]), 0.0f);
    const float iw = fmaxf(fminf(a[3], c[3]) - fmaxf(a[1], c[1]), 0.0f);
    const float inter = ih * iw;
    const float uni = areaA + areaB - inter;
    return (uni > 0.0f) ? inter / uni : 0.0f;
}

__global__ void k_select_nms(const Cand* __restrict__ cand,
                             const float* __restrict__ size_maps,
                             const float* __restrict__ origin,
                             float* __restrict__ planeBox,
                             float* __restrict__ planeScore) {
    const int plane = blockIdx.x;
    const int b = plane / CCH;
    const int t = threadIdx.x;

    __shared__ Cand     scc[CPP];           // TDM destination (30,720 B)
    __shared__ float    rs[THR1];
    __shared__ unsigned rp[THR1];
    __shared__ int      rslot[THR1];
    __shared__ float    selS[NOBJ];
    __shared__ unsigned selP[NOBJ];

    // ---- TDM: DMA the contiguous candidate block (7680 dwords) to LDS.
    // D# per CDNA5 ISA: group0 {count=1, lds_addr,
    // global_addr[56:0], type=2}; group1 {data_size=4B, tensor_dim0=
    // tile_dim0=7680, tensor_dim1=tile_dim1=1, dim0_stride=7680}.
    // TDM ignores EXEC and issues once per wave -> wave 0 only.
    if (t < 32) {
        const unsigned long long gaddr =
            (unsigned long long)(const void*)(cand + (size_t)plane * CPP);
        const unsigned ldsoff = (unsigned)(uintptr_t)(void*)scc; // low 32b = LDS offset
        const unsigned NE = (unsigned)(CPP * (sizeof(Cand) / 4));  // 7680 elements

        u32x4 g0;
        g0[0] = 1u;                                               // count=1
        g0[1] = ldsoff;                                           // lds_addr
        g0[2] = (unsigned)gaddr;                                  // global_addr[31:0]
        g0[3] = ((unsigned)(gaddr >> 32) & 0x01FFFFFFu)           // global_addr[56:32]
              | 0x80000000u;                                      // type=2 ("image")

        u32x8 g1;
        g1[0] = 2u << 16;                  // data_size = 2 (4 bytes)
        g1[1] = (NE & 0xFFFFu) << 16;      // tensor_dim0[15:0]  @ bits[63:48]
        g1[2] = (NE >> 16) | (1u << 16);   // tensor_dim0[31:16] | tensor_dim1=1
        g1[3] = (NE & 0xFFFFu) << 16;      // tile_dim0          @ bits[127:112]
        g1[4] = 1u;                        // tile_dim1=1, tile_dim2=0
        g1[5] = NE;                        // tensor_dim0_stride[31:0]
        g1[6] = 0u;                        // stride hi / dim1_stride lo
        g1[7] = 0u;                        // dim1_stride hi

        asm volatile("tensor_load_to_lds %0, %1" :: "s"(g0), "s"(g1) : "memory");
#if __has_builtin(__builtin_amdgcn_s_wait_tensorcnt)
        __builtin_amdgcn_s_wait_tensorcnt(0);
#else
        asm volatile("s_wait_tensorcnt 0x0" ::: "memory");
#endif
    }
    __syncthreads();                        // LDS now holds the candidates

    // 15 rounds of parallel argmax (ties -> lower pixel index)
    for (int k = 0; k < NOBJ; ++k) {
        float bs = -1e30f; unsigned bp = 0xffffffffu; int bslot = -1;
        for (int i = t; i < CPP; i += THR1) {
            const float s = scc[i].s; const unsigned p = scc[i].p;
            if (s > bs || (s == bs && p < bp)) { bs = s; bp = p; bslot = i; }
        }
        rs[t] = bs; rp[t] = bp; rslot[t] = bslot;
        __syncthreads();
        for (int off = THR1 / 2; off > 0; off >>= 1) {
            if (t < off) {
                const float s2 = rs[t + off]; const unsigned p2 = rp[t + off];
                if (s2 > rs[t] || (s2 == rs[t] && p2 < rp[t])) {
                    rs[t] = s2; rp[t] = p2; rslot[t] = rslot[t + off];
                }
            }
            __syncthreads();
        }
        if (t == 0) {
            selS[k] = rs[0]; selP[k] = rp[0];
            if (rslot[0] >= 0) scc[rslot[0]].s = -1e30f;
        }
        __syncthreads();
    }

    if (t == 0) {
        const float ry = origin[b * 2 + 0] / RESIZE;
        const float rx = origin[b * 2 + 1] / RESIZE;
        float bx[NOBJ][4]; float sc[NOBJ];
        for (int k = 0; k < NOBJ; ++k) {
            float s = selS[k]; const unsigned p = selP[k];
            const int y = (int)(p >> 9), x = (int)(p & (WW - 1));
            const size_t si = (((size_t)b * HH + y) * WW + x) * 2;
            const float h = size_maps[si], w = size_maps[si + 1];
            float b0 = fmaxf((float)y - h * 0.5f, 0.0f) * ry;
            float b1 = fmaxf((float)x - w * 0.5f, 0.0f) * rx;
            float b2 = fminf((float)y + h * 0.5f, RESIZE - 1.0f) * ry;
            float b3 = fminf((float)x + w * 0.5f, RESIZE - 1.0f) * rx;
            if (!(s > KP_THR)) { b0 = b1 = b2 = b3 = -1.0f; s = -1.0f; }
            bx[k][0] = b0; bx[k][1] = b1; bx[k][2] = b2; bx[k][3] = b3;
            sc[k] = s;
        }
        // sequential NMS (list already score-descending)
        bool keep[NOBJ];
        for (int i = 0; i < NOBJ; ++i) {
            bool sup = false;
            for (int j = 0; j < i; ++j)
                if (keep[j] && iou_of(bx[i], bx[j]) > IOU_THR) sup = true;
            keep[i] = !sup;
            if (sup) sc[i] = -INFINITY;
        }
        for (int k = 0; k < NOBJ; ++k) {
            planeScore[plane * NOBJ + k] = sc[k];
            for (int q = 0; q < 4; ++q)
                planeBox[(plane * NOBJ + k) * 4 + q] = bx[k][q];
        }
    }
}

// =====================================================================
// Kernel 3: per batch (one wave32 block) — cross-class merge -> bb, then
// relations. The 15x15 sub·obj Gram matrix over K=4 embedding dims is
// computed with a single V_WMMA_F32_16X16X4_F32 (exact shape match).
// =====================================================================
__global__ void __launch_bounds__(32)
k_merge_rel(const float* __restrict__ planeBox,
            const float* __restrict__ planeScore,
            const float* __restrict__ rel_maps,
            const float* __restrict__ embed_maps,
            const float* __restrict__ origin,
            float* __restrict__ out) {
    const int b = blockIdx.x;
    const int lane = threadIdx.x;

    __shared__ float bbS[NOBJ][6];
    __shared__ float emb[16][ECH];     // row 15 zero-padded for WMMA
    __shared__ float relv[NOBJ][RCH];
    __shared__ int   ciS[NOBJ][2];
    __shared__ float nsub[16], nobj[16];
    __shared__ float Smat[16][16];

    // ---- step 1: cross-class top-15 merge (serial, tiny) ----
    if (lane == 0) {
        float fs[CCH * NOBJ]; float fb[CCH * NOBJ][4];
        for (int c = 0; c < CCH; ++c)
            for (int k = 0; k < NOBJ; ++k) {
                const int plane = b * CCH + c, j = c * NOBJ + k;
                fs[j] = planeScore[plane * NOBJ + k];
                for (int q = 0; q < 4; ++q)
                    fb[j][q] = planeBox[(plane * NOBJ + k) * 4 + q];
            }
        bool used[CCH * NOBJ];
        for (int j = 0; j < CCH * NOBJ; ++j) used[j] = false;
        for (int k = 0; k < NOBJ; ++k) {
            int bi = -1; float best = -INFINITY;
            for (int j = 0; j < CCH * NOBJ; ++j)
                if (!used[j] && (bi == -1 || fs[j] > best)) { bi = j; best = fs[j]; }
            used[bi] = true;
            float e[6];
            if (!isinf(best)) {
                e[0] = fb[bi][0]; e[1] = fb[bi][1]; e[2] = fb[bi][2]; e[3] = fb[bi][3];
                e[4] = best; e[5] = (float)(bi / NOBJ);
            } else {
                e[0] = e[1] = e[2] = e[3] = e[4] = e[5] = 0.0f;
            }
            for (int q = 0; q < 6; ++q) {
                float v = e[q];
                if (v == -1.0f) v = INFINITY;        // sentinel -> inf
                bbS[k][q] = v;
                out[(size_t)b * NOBJ * 6 + k * 6 + q] = v;
            }
        }
    }
    __syncthreads();

    // ---- step 2: gather rel/embed at box centers ----
    const float ry = origin[b * 2 + 0] / RESIZE;
    const float rx = origin[b * 2 + 1] / RESIZE;
    if (lane < NOBJ) {
        float cy = (bbS[lane][0] + bbS[lane][2]) * 0.5f; if (isinf(cy)) cy = 0.0f;
        float cx = (bbS[lane][1] + bbS[lane][3]) * 0.5f; if (isinf(cx)) cx = 0.0f;
        cy /= ry; cx /= rx;
        if (isinf(cy)) cy = 0.0f; if (isinf(cx)) cx = 0.0f;
        const int ci0 = (int)(cy + 0.5f), ci1 = (int)(cx + 0.5f);
        ciS[lane][0] = ci0; ciS[lane][1] = ci1;
        const int gy = min(max(ci0, 0), HH - 1), gx = min(max(ci1, 0), WW - 1);
        const size_t pix = ((size_t)b * HH + gy) * WW + gx;
        for (int q = 0; q < RCH; ++q) relv[lane][q] = rel_maps[pix * RCH + q];
        float nS = 0.0f, nO = 0.0f;
        for (int q = 0; q < ECH; ++q) {
            const float v = embed_maps[pix * ECH + q];
            emb[lane][q] = v;
            if (q < ECH / 2) nS += v * v; else nO += v * v;
        }
        nsub[lane] = nS; nobj[lane] = nO;
    } else if (lane == NOBJ) {
        for (int q = 0; q < ECH; ++q) emb[NOBJ][q] = 0.0f;
        nsub[NOBJ] = 0.0f; nobj[NOBJ] = 0.0f;
    }
    __syncthreads();

    // ---- step 3: Gram matrix S = sub (16x4) x obj^T (4x16) via WMMA ----
    // A layout (16x4 f32): lanes 0-15 -> M=lane, K={0,1}; lanes 16-31 -> M=lane-16, K={2,3}
    // B layout (4x16 f32): lanes 0-15 -> N=lane, K={0,1}; lanes 16-31 -> N=lane-16, K={2,3}
    {
        const int r = lane & 15;
        const int koff = (lane >> 4) * 2;
        v2f av = { emb[r][koff],     emb[r][koff + 1] };
        v2f bv = { emb[r][4 + koff], emb[r][5 + koff] };
        v8f cz = {0.0f, 0.0f, 0.0f, 0.0f, 0.0f, 0.0f, 0.0f, 0.0f};
        v8f dm;
#if __has_builtin(__builtin_amdgcn_wmma_f32_16x16x4_f32)
        dm = __builtin_amdgcn_wmma_f32_16x16x4_f32(
            /*neg_a=*/false, av, /*neg_b=*/false, bv,
            /*c_mod=*/(short)0, cz, /*reuse_a=*/false, /*reuse_b=*/false);
#else
        for (int q = 0; q < 8; ++q) dm[q] = 0.0f;
#endif
        // D layout: VGPR q holds row (q + 8*(lane>=16)), col = lane&15
        const int rb = (lane >= 16) ? 8 : 0;
        const int col = lane & 15;
#pragma unroll
        for (int q = 0; q < 8; ++q) Smat[rb + q][col] = dm[q];
    }
    __syncthreads();

    // ---- step 4: distances, masks, top-3 nearest, rel argmax, output ----
    if (lane < NOBJ) {
        const int i = lane;
        const float mi0 = (ciS[i][0] != 0) ? 1.0f : 0.0f;
        const float mi1 = (ciS[i][1] != 0) ? 1.0f : 0.0f;
        float dd[NOBJ];
        for (int j = 0; j < NOBJ; ++j) {
            const float mj0 = (ciS[j][0] != 0) ? 1.0f : 0.0f;
            const float mj1 = (ciS[j][1] != 0) ? 1.0f : 0.0f;
            const float m2 = (mi0 * mj0 + mi1 * mj1) * 0.5f;
            const float dsq = nsub[i] + nobj[j] - 2.0f * Smat[i][j];
            float d = sqrtf(fmaxf(dsq, 0.0f)) * m2;
            if (d == 0.0f) d = INFINITY;
            dd[j] = d;
        }
        int pidx[KPAIR];
        bool udd[NOBJ];
        for (int j = 0; j < NOBJ; ++j) udd[j] = false;
        for (int k = 0; k < KPAIR; ++k) {           // 3 smallest, ties -> lowest idx
            int bj = -1; float best = INFINITY;
            for (int j = 0; j < NOBJ; ++j)
                if (!udd[j] && (bj == -1 || dd[j] < best)) { bj = j; best = dd[j]; }
            udd[bj] = true; pidx[k] = bj;
        }
        // softmax is monotonic -> argmax of raw logits
        int ai = 0; float avv = relv[i][0];
        for (int q = 1; q < RCH; ++q)
            if (relv[i][q] > avv) { avv = relv[i][q]; ai = q; }
        const float rmask = mi0;

        float* ro = out + (size_t)BATCH * NOBJ * 6 + (size_t)b * NOBJ * KPAIR * 3;
        for (int k = 0; k < KPAIR; ++k) {
            const int row = k * NOBJ + i;
            float v0 = (float)(i + 1) * rmask;
            float v1 = (float)(ai + 1) * rmask;
            float v2 = (float)(pidx[k] + 1) * rmask;
            if (v0 == 0.0f) v0 = INFINITY;
            if (v1 == 0.0f) v1 = INFINITY;
            if (v2 == 0.0f) v2 = INFINITY;
            ro[row * 3 + 0] = v0 - 1.0f;
            ro[row * 3 + 1] = v1 - 1.0f;
            ro[row * 3 + 2] = v2 - 1.0f;
        }
    }
}

// =====================================================================
extern "C" void kernel_launch(void* const* d_in, const int* in_sizes, int n_in,
                              void* d_out, int out_size, void* d_ws, size_t ws_size,
                              hipStream_t stream) {
    const float* hm   = (const float*)d_in[0];   // (16,512,512,3)
    const float* szm  = (const float*)d_in[1];   // (16,512,512,2)
    const float* rel  = (const float*)d_in[2];   // (16,512,512,7)
    const float* embm = (const float*)d_in[3];   // (16,512,512,8)
    const float* org  = (const float*)d_in[4];   // (16,2)
    float* out = (float*)d_out;                  // 1440 bb + 2160 rels

    char* ws = (char*)d_ws;
    Cand* cand = (Cand*)ws;
    const size_t candBytes = sizeof(Cand) * (size_t)PLANES * CPP;   // ~1.47 MB
    float* planeBox   = (float*)(ws + candBytes);                   // 48*15*4
    float* planeScore = planeBox + PLANES * NOBJ * 4;               // 48*15

    k_peaks<<<PLANES, THR1, 0, stream>>>(hm, cand);
    k_select_nms<<<PLANES, THR1, 0, stream>>>(cand, szm, org, planeBox, planeScore);
    k_merge_rel<<<BATCH, 32, 0, stream>>>(planeBox, planeScore, rel, embm, org, out);
    (void)in_sizes; (void)n_in; (void)out_size; (void)ws_size;
}